// Graph_Attention_Hierarchy_Triples_29523605192708
// MI455X (gfx1250) — compile-verified
//
#include <hip/hip_runtime.h>

#define B_ 32
#define G_ 64
#define T_ 64
#define E_ 256
#define H_ 1024

typedef float v2f __attribute__((ext_vector_type(2)));
typedef float v8f __attribute__((ext_vector_type(8)));

// ---------------------------------------------------------------------------
// Kernel 0: zero the c_hierarchical region of d_out (atomic accumulation target)
// ---------------------------------------------------------------------------
__global__ void zero_c_kernel(float* __restrict__ c) {
    c[blockIdx.x * blockDim.x + threadIdx.x] = 0.0f;
}

// ---------------------------------------------------------------------------
// Kernel 1: intermediate[B,E] = dh[B,H] @ W[E,H]^T using v_wmma_f32_16x16x4_f32
// One wave (32 threads) per 16x16 output tile; K loop in steps of 4.
// A (16x4 f32): lane<16 -> M=lane, K={k0,k0+1}; lane>=16 -> M=lane-16, K={k0+2,k0+3}
// B (4x16 f32): lane<16 -> N=lane, K={k0,k0+1}; lane>=16 -> N=lane-16, K={k0+2,k0+3}
// Since B[k][n] = W[n][k] (W row-major [E,H]), both A and B are contiguous
// float2 loads along the K axis.
// ---------------------------------------------------------------------------
__global__ void __launch_bounds__(32) wmma_gemm_kernel(const float* __restrict__ dh,
                                                       const float* __restrict__ W,
                                                       float* __restrict__ inter) {
    const int lane  = threadIdx.x;        // 0..31, exactly one wave32
    const int mTile = blockIdx.x >> 4;    // 0..1   (M = 32)
    const int nTile = blockIdx.x & 15;    // 0..15  (N = 256)
    const int rc    = lane & 15;
    const int khalf = (lane >> 4) << 1;   // 0 or 2

    const float* aBase = dh + (size_t)(mTile * 16 + rc) * H_ + khalf;
    const float* bBase = W  + (size_t)(nTile * 16 + rc) * H_ + khalf;

    v8f acc = {0.f, 0.f, 0.f, 0.f, 0.f, 0.f, 0.f, 0.f};
    for (int k0 = 0; k0 < H_; k0 += 4) {
        v2f a = *(const v2f*)(aBase + k0);
        v2f b = *(const v2f*)(bBase + k0);
        acc = __builtin_amdgcn_wmma_f32_16x16x4_f32(
            /*neg_a=*/false, a, /*neg_b=*/false, b,
            /*c_mod=*/(short)0, acc, /*reuse_a=*/false, /*reuse_b=*/false);
    }

    // C/D layout: n = lane&15 ; m = v + 8*(lane>=16)
    const int n     = nTile * 16 + (lane & 15);
    const int mBase = mTile * 16 + ((lane >> 4) << 3);
#pragma unroll
    for (int v = 0; v < 8; ++v) {
        inter[(size_t)(mBase + v) * E_ + n] = acc[v];
    }
}

// ---------------------------------------------------------------------------
// Kernel 2: fused beta / masked-softmax / alpha / c for one (b,g) tile.
// 256 threads (8 wave32s) per block; one block per (b,g).
// ---------------------------------------------------------------------------
__global__ void __launch_bounds__(256) fused_attn_kernel(
    const float* __restrict__ emb,        // [B,G,T,E]
    const float* __restrict__ inter,      // [B,E]
    const float* __restrict__ alpha_top,  // [B,G]
    const int*   __restrict__ mask,       // [B,G,T]
    float*       __restrict__ c_out,      // [B,E]  (atomic accumulation)
    float*       __restrict__ alpha_out)  // [B,G,T]
{
    __shared__ float s_inter[E_];
    __shared__ float s_beta[T_];
    __shared__ float s_a[T_];
    __shared__ int   s_mask[T_];

    const int bg  = blockIdx.x;
    const int b   = bg / G_;
    const int tid = threadIdx.x;
    const float* tile = emb + (size_t)bg * T_ * E_;

    s_inter[tid] = inter[b * E_ + tid];
    if (tid < T_) s_mask[tid] = mask[(size_t)bg * T_ + tid];
    __syncthreads();

    // ---- Pass 1: beta[t] = dot(tile[t,:], inter[b,:]) ----
    // 4 threads per row t; each streams 64 contiguous floats as float4.
    const int t_row = tid >> 2;
    const int q     = tid & 3;
    const float4* rp = (const float4*)(tile + (size_t)t_row * E_ + q * 64);
    const float4* ip = (const float4*)(&s_inter[q * 64]);
    float partial = 0.f;
#pragma unroll
    for (int j = 0; j < 16; ++j) {
        float4 ev = rp[j];
        float4 iv = ip[j];
        partial += ev.x * iv.x + ev.y * iv.y + ev.z * iv.z + ev.w * iv.w;
    }
    // quad lanes are consecutive within a wave32: xor-reduce over {1,2}
    partial += __shfl_xor(partial, 1, 32);
    partial += __shfl_xor(partial, 2, 32);
    if (q == 0) s_beta[t_row] = partial;
    __syncthreads();

    // ---- masked softmax over t, alpha = alpha_top * p (invalid -> 0) ----
    if (tid < T_) {
        float mx = -3.402823466e+38f;
        for (int t = 0; t < T_; ++t)
            if (s_mask[t]) mx = fmaxf(mx, s_beta[t]);
        float sum = 0.f;
        for (int t = 0; t < T_; ++t)
            if (s_mask[t]) sum += __expf(s_beta[t] - mx);
        const float inv  = (sum > 0.f) ? (1.0f / sum) : 0.f;
        const float atop = alpha_top[bg];
        const float a_t  = s_mask[tid] ? (__expf(s_beta[tid] - mx) * inv * atop) : 0.f;
        s_a[tid] = a_t;
        alpha_out[(size_t)bg * T_ + tid] = a_t;
    }
    __syncthreads();

    // ---- Pass 2: c[b,e] += sum_t a[t] * tile[t,e]  (e = tid) ----
    // Re-reads the 64KB tile just touched (WGP$/L2 hit); fully coalesced.
    float accum = 0.f;
#pragma unroll 8
    for (int t = 0; t < T_; ++t)
        accum += s_a[t] * tile[(size_t)t * E_ + tid];
    atomicAdd(&c_out[b * E_ + tid], accum);
}

// ---------------------------------------------------------------------------
extern "C" void kernel_launch(void* const* d_in, const int* in_sizes, int n_in,
                              void* d_out, int out_size, void* d_ws, size_t ws_size,
                              hipStream_t stream) {
    const float* dh   = (const float*)d_in[0];  // [B,H]
    const float* atop = (const float*)d_in[1];  // [B,G]
    const float* emb  = (const float*)d_in[2];  // [B,G,T,E]
    const float* W    = (const float*)d_in[3];  // [E,H]
    const int*   mask = (const int*)d_in[4];    // [B,G,T]

    float* out       = (float*)d_out;
    float* c_out     = out;                 // [B,E]   first output
    float* alpha_out = out + B_ * E_;       // [B,G,T] second output
    float* inter     = (float*)d_ws;        // [B,E]   scratch

    zero_c_kernel<<<B_, E_, 0, stream>>>(c_out);
    wmma_gemm_kernel<<<(B_ / 16) * (E_ / 16), 32, 0, stream>>>(dh, W, inter);
    fused_attn_kernel<<<B_ * G_, 256, 0, stream>>>(emb, inter, atop, mask, c_out, alpha_out);
}